// GlobalChannelCovariancePooling3D_25031069401648
// MI455X (gfx1250) — compile-verified
//
#include <hip/hip_runtime.h>

typedef __attribute__((ext_vector_type(2))) float v2f;
typedef __attribute__((ext_vector_type(8))) float v8f;
typedef __attribute__((address_space(3))) float lds_f32;

#define B_        8
#define D_        64
#define K_        131072      // C*H*W = 32*64*64 (memory-order reduction axis)
#define CHUNKS    64          // K-chunks per batch -> 512 workgroups total
#define KC_PER_WG (K_ / CHUNKS)   // 2048
#define TILE_K    64
#define ITERS     (KC_PER_WG / TILE_K)  // 32
#define LDS_S     68          // 64+4 pad: stride mod 64 == 4 -> conflict-free ds_load_b64
#define LDS_BUF_F (64 * LDS_S)          // floats per buffer (17408 B)
#define NTHREADS  128         // 4 waves (wave32)

// CDNA5 async global->LDS DMA (no data VGPRs, tracked by ASYNCcnt).
__device__ __forceinline__ void async_load_b128(uint32_t lds_addr, const float* gp) {
    asm volatile("global_load_async_to_lds_b128 %0, %1, off"
                 :: "v"(lds_addr), "v"(gp)
                 : "memory");
}
__device__ __forceinline__ void wait_asynccnt0() {
    asm volatile("s_wait_asynccnt 0" ::: "memory");
}

// ---------------------------------------------------------------------------
// Main kernel: per-workgroup partial Gram matrix (64x64) + column sums,
// accumulated into workspace with hardware f32 atomics.
// ---------------------------------------------------------------------------
__global__ __launch_bounds__(NTHREADS)
void cov_partial_kernel(const float* __restrict__ x,
                        float* __restrict__ P,   // [B][64][64] raw cross sums
                        float* __restrict__ S)   // [B][64]     column sums
{
    const int wg    = blockIdx.x;
    const int b     = wg / CHUNKS;
    const int chunk = wg % CHUNKS;
    const int tid   = threadIdx.x;
    const int lane  = tid & 31;
    const int wave  = tid >> 5;

    const float* __restrict__ Xb = x + (size_t)b * D_ * K_;

    __shared__ float lds[2 * LDS_BUF_F];   // double-buffered, 34816 B
    const uint32_t lds_base = (uint32_t)(uintptr_t)(lds_f32*)&lds[0];

    v8f zero = {};
    v8f acc[16];
#pragma unroll
    for (int i = 0; i < 16; ++i) acc[i] = zero;
    float ps[4] = {0.f, 0.f, 0.f, 0.f};

    const int kbase0 = chunk * KC_PER_WG;
    const int r0 = tid >> 4;          // 0..7  (base row of this thread's loads)
    const int c0 = (tid & 15) << 2;   // 0..60 step 4 (float col of this thread)

    // Issue one 64x64 tile's async loads: 8 x b128 per thread, coalesced
    // 256 B per row segment; per-lane LDS addresses honor the padded stride.
    auto issue_tile = [&](int kb, int buf) {
#pragma unroll
        for (int i = 0; i < 8; ++i) {
            const int row = r0 + 8 * i;
            const float* gp = Xb + (size_t)row * K_ + (size_t)(kb + c0);
            const uint32_t la =
                lds_base + (uint32_t)((buf * LDS_BUF_F + row * LDS_S + c0) * 4);
            async_load_b128(la, gp);
        }
    };

    issue_tile(kbase0, 0);   // prologue: tile 0 -> buffer 0

    const int rowl = lane & 15;
    const int khal = (lane >> 4) << 1;   // 0 or 2

    for (int t = 0; t < ITERS; ++t) {
        wait_asynccnt0();     // this wave's tile-t loads landed
        __syncthreads();      // all waves landed + all done reading buf[(t+1)&1]

        if (t + 1 < ITERS)
            issue_tile(kbase0 + (t + 1) * TILE_K, (t + 1) & 1);

        const float* buf = &lds[(t & 1) * LDS_BUF_F];

        // Each wave consumes its private 16 k-columns: [wave*16, wave*16+16).
#pragma unroll
        for (int kk = 0; kk < 16; kk += 4) {
            const int kcol = wave * 16 + kk + khal;
            // Gram trick: A-fragment (16x4) and B-fragment (4x16) of the same
            // 16-row tile have identical lane layouts -> 4 loads, 16 WMMAs.
            v2f frag[4];
#pragma unroll
            for (int r = 0; r < 4; ++r) {
                frag[r] = *(const v2f*)&buf[(r * 16 + rowl) * LDS_S + kcol];
                ps[r] += frag[r].x + frag[r].y;   // column-sum partials (VALU)
            }
#pragma unroll
            for (int r = 0; r < 4; ++r) {
#pragma unroll
                for (int c = 0; c < 4; ++c) {
                    acc[r * 4 + c] = __builtin_amdgcn_wmma_f32_16x16x4_f32(
                        false, frag[r], false, frag[c],
                        (short)0, acc[r * 4 + c], false, false);
                }
            }
        }
    }

    // ---- epilogue: accumulate into workspace ----
    // Column sums: lanes L and L+16 hold the two k-halves of row r*16+(L&15).
#pragma unroll
    for (int r = 0; r < 4; ++r) {
        float v = ps[r];
        v += __shfl_xor(v, 16, 32);
        if (lane < 16)
            unsafeAtomicAdd(&S[b * D_ + r * 16 + lane], v);
    }
    // C/D 16x16 layout: lanes 0-15 -> (M=v, N=lane); lanes 16-31 -> (M=v+8, N=lane-16)
    const int ncol  = lane & 15;
    const int mbase = (lane >> 4) << 3;
#pragma unroll
    for (int r = 0; r < 4; ++r) {
#pragma unroll
        for (int c = 0; c < 4; ++c) {
#pragma unroll
            for (int v = 0; v < 8; ++v) {
                int d = r * 16 + mbase + v;
                int e = c * 16 + ncol;
                unsafeAtomicAdd(&P[((size_t)b * D_ + d) * D_ + e], acc[r * 4 + c][v]);
            }
        }
    }
}

// ---------------------------------------------------------------------------
// Finalize: cov = P/N - (S/N)(S/N)^T
// ---------------------------------------------------------------------------
__global__ __launch_bounds__(256)
void cov_finalize_kernel(const float* __restrict__ P,
                         const float* __restrict__ S,
                         float* __restrict__ out)
{
    int i = blockIdx.x * 256 + threadIdx.x;    // over B*D*D = 32768
    if (i >= B_ * D_ * D_) return;
    int b  = i / (D_ * D_);
    int de = i - b * (D_ * D_);
    int d  = de >> 6;
    int e  = de & 63;
    const float invN = 1.0f / (float)K_;
    float mu_d = S[b * D_ + d] * invN;
    float mu_e = S[b * D_ + e] * invN;
    out[i] = P[i] * invN - mu_d * mu_e;
}

// ---------------------------------------------------------------------------
extern "C" void kernel_launch(void* const* d_in, const int* in_sizes, int n_in,
                              void* d_out, int out_size, void* d_ws, size_t ws_size,
                              hipStream_t stream)
{
    (void)in_sizes; (void)n_in; (void)out_size; (void)ws_size;
    const float* x = (const float*)d_in[0];
    float* out = (float*)d_out;

    float* P = (float*)d_ws;                     // B*D*D floats
    float* S = P + (size_t)B_ * D_ * D_;         // B*D floats
    size_t zbytes = (size_t)(B_ * D_ * D_ + B_ * D_) * sizeof(float);

    hipMemsetAsync(d_ws, 0, zbytes, stream);
    cov_partial_kernel<<<dim3(B_ * CHUNKS), dim3(NTHREADS), 0, stream>>>(x, P, S);
    cov_finalize_kernel<<<dim3((B_ * D_ * D_ + 255) / 256), dim3(256), 0, stream>>>(P, S, out);
}